// ScaleDotProductAttention_77094662963310
// MI455X (gfx1250) — compile-verified
//
#include <hip/hip_runtime.h>
#include <stdint.h>

typedef __attribute__((ext_vector_type(2))) float v2f;
typedef __attribute__((ext_vector_type(8))) float v8f;

// Problem constants (from reference): B=4, H=16, L=2048, D=64
constexpr int Bn = 4;
constexpr int Hn = 16;
constexpr int Ln = 2048;
constexpr int Dn = 64;

// Dynamic LDS layout (floats):
//   [0, 16*2048)          : S  — 16 x 2048 score tile (raw -> exp -> normalized)
//   [16*2048, +128)       : red — 16 rows x 8 partial reductions
//   [.. +16)              : rowstat — per-row max, then per-row sum
constexpr size_t kSmemFloats = (size_t)16 * Ln + 128 + 16;

__global__ __launch_bounds__(128) void
sdpa_wmma_f32_kernel(const float* __restrict__ q, const float* __restrict__ k,
                     const float* __restrict__ v, const uint8_t* __restrict__ mask,
                     float* __restrict__ out, float* __restrict__ score) {
  extern __shared__ float smem[];
  float* S       = smem;                       // 16 x 2048
  float* red     = smem + (size_t)16 * Ln;     // 128
  float* rowstat = red + 128;                  // 16

  const int bh = blockIdx.x >> 7;   // (b*H + h), 0..63
  const int qt = blockIdx.x & 127;  // query tile within head
  const int b  = bh >> 4;
  const int q0 = qt * 16;

  const int tid  = threadIdx.x;     // 0..127
  const int wave = tid >> 5;        // 0..3
  const int lane = tid & 31;
  const int nrow = lane & 15;             // M (A/C row) or N (B/C col) index
  const int dsel = (lane >> 4) * 2;       // per-lane K sub-offset for A/B frags
  const int rbase = (lane >> 4) * 8;      // C/D: VGPR r holds row rbase + r

  const float* qh = q + (size_t)bh * Ln * Dn;
  const float* kh = k + (size_t)bh * Ln * Dn;
  const float* vh = v + (size_t)bh * Ln * Dn;

  // ---------------- Phase 1: raw S = Q * K^T --------------------------------
  // f32 A-matrix 16x4 layout: lane (M = lane&15) holds K = 4s + dsel, 4s+dsel+1
  v2f afrag[16];
  {
    const float* qrow = qh + (size_t)(q0 + nrow) * Dn + dsel;
#pragma unroll
    for (int s = 0; s < 16; ++s) afrag[s] = *(const v2f*)(qrow + 4 * s);
  }
  for (int j = wave; j < Ln / 16; j += 4) {
    // B = K^T (4x16 per step): lane (N = lane&15) holds Kmat[j*16+N][4s+dsel..+1]
    const float* krow = kh + (size_t)(j * 16 + nrow) * Dn + dsel;
    v8f acc = {};
#pragma unroll
    for (int s = 0; s < 16; ++s) {
      v2f bfrag = *(const v2f*)(krow + 4 * s);
      acc = __builtin_amdgcn_wmma_f32_16x16x4_f32(false, afrag[s], false, bfrag,
                                                  (short)0, acc, false, false);
    }
#pragma unroll
    for (int r = 0; r < 8; ++r)
      S[(size_t)(rbase + r) * Ln + j * 16 + nrow] = acc[r];
  }
  __syncthreads();

  // ---------------- Phase 2: scale, mask, softmax, emit score ---------------
  {
    const int r  = tid >> 3;      // row 0..15
    const int cg = tid & 7;       // 8 threads per row
    const int c0 = cg * 256;      // 256 contiguous columns each
    const uint8_t* mrow = mask + ((size_t)b * Ln + q0 + r) * Ln;
    float* Sr = S + (size_t)r * Ln;

    float m = -3.0e38f;
    for (int i = 0; i < 256; ++i) {
      const int c = c0 + i;
      float val = Sr[c] * 0.125f;              // 1/sqrt(64)
      if (mrow[c]) val = -1.0e-10f;            // faithful masked_fill(-1e-10)
      Sr[c] = val;
      m = fmaxf(m, val);
    }
    red[tid] = m;
    __syncthreads();
    if (cg == 0) {
      float mm = red[r * 8];
#pragma unroll
      for (int t = 1; t < 8; ++t) mm = fmaxf(mm, red[r * 8 + t]);
      rowstat[r] = mm;
    }
    __syncthreads();
    const float mx = rowstat[r];

    float sum = 0.0f;
    for (int i = 0; i < 256; ++i) {
      const int c = c0 + i;
      const float e = __expf(Sr[c] - mx);
      Sr[c] = e;
      sum += e;
    }
    red[tid] = sum;
    __syncthreads();
    if (cg == 0) {
      float ss = red[r * 8];
#pragma unroll
      for (int t = 1; t < 8; ++t) ss += red[r * 8 + t];
      rowstat[r] = ss;
    }
    __syncthreads();
    const float inv = 1.0f / rowstat[r];

    float* scRow = score + ((size_t)bh * Ln + q0 + r) * Ln;
    for (int i = 0; i < 256; ++i) {
      const int c = c0 + i;
      const float p = Sr[c] * inv;
      Sr[c]    = p;
      scRow[c] = p;   // dominant 1.07 GB store, per-thread contiguous, lanes 256 floats apart
    }
  }
  __syncthreads();

  // ---------------- Phase 3: out = P * V (each wave: 16 output cols) -------
  {
    const int n0 = wave * 16;
    const float* vcol = vh + n0 + nrow;              // column (n0 + N) of V
    const float* Sa = S + (size_t)nrow * Ln + dsel;  // A row M = lane&15
    v8f acc = {};
#pragma unroll 8
    for (int kk = 0; kk < Ln; kk += 4) {
      v2f af = *(const v2f*)(Sa + kk);               // LDS b64 load
      v2f bf;
      bf.x = vcol[(size_t)(kk + dsel) * Dn];         // V[k][n], L2-resident
      bf.y = vcol[(size_t)(kk + dsel + 1) * Dn];
      acc = __builtin_amdgcn_wmma_f32_16x16x4_f32(false, af, false, bf,
                                                  (short)0, acc, false, false);
    }
    float* oh = out + ((size_t)bh * Ln + q0) * Dn + n0 + nrow;
#pragma unroll
    for (int r = 0; r < 8; ++r)
      oh[(size_t)(rbase + r) * Dn] = acc[r];
  }
}

extern "C" void kernel_launch(void* const* d_in, const int* in_sizes, int n_in,
                              void* d_out, int out_size, void* d_ws, size_t ws_size,
                              hipStream_t stream) {
  const float*   q    = (const float*)d_in[0];
  const float*   k    = (const float*)d_in[1];
  const float*   v    = (const float*)d_in[2];
  const uint8_t* mask = (const uint8_t*)d_in[3];  // jnp.bool_ -> 1 byte/elem

  float* out   = (float*)d_out;                               // [B,H,L,D]
  float* score = out + (size_t)Bn * Hn * Ln * Dn;             // [B,H,L,L]

  const size_t shmem = kSmemFloats * sizeof(float);           // ~128.6 KB
  // Host-side attribute set (not a stream op — safe under graph capture).
  hipFuncSetAttribute((const void*)sdpa_wmma_f32_kernel,
                      hipFuncAttributeMaxDynamicSharedMemorySize, (int)shmem);

  const int grid = Bn * Hn * (Ln / 16);  // 8192 blocks
  sdpa_wmma_f32_kernel<<<grid, 128, shmem, stream>>>(q, k, v, mask, out, score);
}